// STOnco_Classifier_80247168958755
// MI455X (gfx1250) — compile-verified
//
#include <hip/hip_runtime.h>
#include <math.h>

// ---------------------------------------------------------------------------
// STOnco GNN classifier for MI455X (gfx1250), fp32 via V_WMMA_F32_16X16X4_F32.
// v2: weights staged in LDS (320KB/WGP), batched B-fragment ds_loads +
// double-buffered A loads so WMMAs issue back-to-back instead of
// load->wait->wmma serialization seen in v1 disassembly.
// ---------------------------------------------------------------------------

typedef __attribute__((ext_vector_type(2))) float v2f;
typedef __attribute__((ext_vector_type(8))) float v8f;

#define N_NODES 50000
#define N_EDGES 800000
#define HDIM    128
#define LN_EPS  1e-5f

// ---------------------------------------------------------------- utilities
__global__ void zero_f32(float* __restrict__ p, int n) {
  int i = blockIdx.x * blockDim.x + threadIdx.x;
  int stride = gridDim.x * blockDim.x;
  for (; i < n; i += stride) p[i] = 0.0f;
}

// sum_w[dst] += ew  (same for every layer -> computed once)
__global__ void sumw_kernel(const int* __restrict__ dst, const float* __restrict__ ew,
                            float* __restrict__ sumw, int E) {
  int e = blockIdx.x * blockDim.x + threadIdx.x;
  if (e < E) atomicAdd(&sumw[dst[e]], ew[e]);
}

__global__ void invw_kernel(const float* __restrict__ sumw, float* __restrict__ invw, int n) {
  int i = blockIdx.x * blockDim.x + threadIdx.x;
  if (i < n) invw[i] = 1.0f / fmaxf(sumw[i], 1.0f);
}

// agg[dst,:] += ew[e] * h[src,:]   one wave per edge, 4 channels per lane
__global__ void scatter_kernel(const float* __restrict__ h, const float* __restrict__ ew,
                               const int* __restrict__ src, const int* __restrict__ dst,
                               float* __restrict__ agg, int E) {
  long long gid = (long long)blockIdx.x * blockDim.x + threadIdx.x;
  int e = (int)(gid >> 5);
  if (e >= E) return;
  int c = ((int)gid & 31) * 4;
  int s = src[e], d = dst[e];
  float w = ew[e];
  const float4 hv = *(const float4*)(h + (long long)s * HDIM + c);
  float* o = agg + (long long)d * HDIM + c;
  atomicAdd(o + 0, w * hv.x);
  atomicAdd(o + 1, w * hv.y);
  atomicAdd(o + 2, w * hv.z);
  atomicAdd(o + 3, w * hv.w);
}

// ------------------------------------------------- conv layer: fused WMMA GEMM
// out = relu( (agg*invw) @ Wn^T + hin @ Wr^T + b ); accumulates sum/sumsq for
// graph-mode LayerNorm into stats[0]/stats[1].
// Block = 256 threads = 8 waves; both 128x128 weight matrices staged in LDS
// (128KB of the 320KB/WGP); wave w covers rows [bx*128+w*16, +16), all 128 cols.
__global__ __launch_bounds__(256)
void conv_gemm_kernel(const float* __restrict__ agg, const float* __restrict__ invw,
                      const float* __restrict__ hin,
                      const float* __restrict__ Wn, const float* __restrict__ Wr,
                      const float* __restrict__ bias,
                      float* __restrict__ out, float* __restrict__ stats, int n) {
  __shared__ float sW[2 * HDIM * HDIM];          // 128 KB
  float* sWn = sW;
  float* sWr = sW + HDIM * HDIM;
  for (int i = threadIdx.x; i < (HDIM * HDIM) / 4; i += 256) {
    ((float4*)sWn)[i] = ((const float4*)Wn)[i];
    ((float4*)sWr)[i] = ((const float4*)Wr)[i];
  }
  __syncthreads();

  const int lane = threadIdx.x & 31;
  const int wave = threadIdx.x >> 5;
  const int rb   = blockIdx.x * 128 + wave * 16;
  const int lrow = lane & 15;             // A-frag row / B-frag col  (ISA 7.12.2)
  const int koff = (lane >> 4) * 2;       // K split between lane halves
  int arow = rb + lrow;
  if (arow >= n) arow = n - 1;            // clamp (stores are masked below)
  const float iw = invw[arow];

  const float* aN = agg + (long long)arow * HDIM + koff;
  const float* aR = hin + (long long)arow * HDIM + koff;
  const float* bN = sWn + lrow * HDIM + koff;
  const float* bR = sWr + lrow * HDIM + koff;

  v8f acc[8] = {};
  // software pipeline the two global A loads across k iterations
  v2f an = *(const v2f*)aN;
  v2f ar = *(const v2f*)aR;

#pragma unroll 2
  for (int k = 0; k < HDIM; k += 4) {
    v2f an_nxt = an, ar_nxt = ar;
    if (k + 4 < HDIM) {
      an_nxt = *(const v2f*)(aN + k + 4);
      ar_nxt = *(const v2f*)(aR + k + 4);
    }
    v2f cn = an;
    cn.x *= iw; cn.y *= iw;               // fused weighted-mean normalization

    v2f bn[8], br[8];                     // batch all 16 LDS loads first...
#pragma unroll
    for (int t = 0; t < 8; ++t) {
      bn[t] = *(const v2f*)(bN + t * 16 * HDIM + k);
      br[t] = *(const v2f*)(bR + t * 16 * HDIM + k);
    }
#pragma unroll
    for (int t = 0; t < 8; ++t)           // ...then 16 WMMAs back-to-back
      acc[t] = __builtin_amdgcn_wmma_f32_16x16x4_f32(false, cn, false, bn[t],
                                                     (short)0, acc[t], false, false);
#pragma unroll
    for (int t = 0; t < 8; ++t)
      acc[t] = __builtin_amdgcn_wmma_f32_16x16x4_f32(false, ar, false, br[t],
                                                     (short)0, acc[t], false, false);
    an = an_nxt;
    ar = ar_nxt;
  }

  float s = 0.0f, ss = 0.0f;
#pragma unroll
  for (int t = 0; t < 8; ++t) {
    const int col = t * 16 + lrow;
    const float b = bias[col];
#pragma unroll
    for (int v = 0; v < 8; ++v) {
      const int row = rb + v + ((lane >> 4) * 8);   // C/D layout (ISA 7.12.2)
      float z = fmaxf(acc[t][v] + b, 0.0f);
      if (row < n) {
        out[(long long)row * HDIM + col] = z;
        s += z;
        ss += z * z;
      }
    }
  }
  // wave32 reduction for LayerNorm statistics
  for (int off = 16; off > 0; off >>= 1) {
    s  += __shfl_down(s, off, 32);
    ss += __shfl_down(ss, off, 32);
  }
  if (lane == 0) {
    atomicAdd(&stats[0], s);
    atomicAdd(&stats[1], ss);
  }
}

__global__ void ln_finalize(float* __restrict__ stats, float count) {
  float mean = stats[0] / count;
  float var  = fmaxf(stats[1] / count - mean * mean, 0.0f);
  stats[2] = mean;
  stats[3] = 1.0f / (sqrtf(var) + LN_EPS);
}

__global__ void ln_apply(float* __restrict__ h, const float* __restrict__ stats,
                         const float* __restrict__ g, const float* __restrict__ b,
                         int total) {
  int i = blockIdx.x * blockDim.x + threadIdx.x;
  if (i >= total) return;
  int c = i & (HDIM - 1);
  h[i] = (h[i] - stats[2]) * stats[3] * g[c] + b[c];
}

// --------------------------------------- classifier: WMMA GEMM + BN(eval)+ReLU
// out[n,NOUT] = relu( BN( in[n,K] @ W[NOUT,K]^T + bias ) )
// Block = 128 threads = 4 waves stacked in M; the block's 16-column weight
// tile (16xK floats) staged in LDS; grid.y tiles NOUT/16.
template <int K, int NOUT>
__global__ __launch_bounds__(128)
void clf_gemm_kernel(const float* __restrict__ in, const float* __restrict__ W,
                     const float* __restrict__ bias,
                     const float* __restrict__ bng, const float* __restrict__ bnb,
                     const float* __restrict__ bnm, const float* __restrict__ bnv,
                     float* __restrict__ out, int n) {
  __shared__ float sW[16 * K];
  const int cb = blockIdx.y * 16;
  for (int i = threadIdx.x; i < (16 * K) / 4; i += 128)
    ((float4*)sW)[i] = ((const float4*)(W + (long long)cb * K))[i];
  __syncthreads();

  const int lane = threadIdx.x & 31;
  const int wave = threadIdx.x >> 5;
  const int rb   = blockIdx.x * 64 + wave * 16;
  const int lrow = lane & 15;
  const int koff = (lane >> 4) * 2;
  int arow = rb + lrow;
  if (arow >= n) arow = n - 1;

  const float* ap = in + (long long)arow * K + koff;
  const float* wp = sW + lrow * K + koff;

  v8f acc = {};
  v2f a = *(const v2f*)ap;
#pragma unroll 4
  for (int k = 0; k < K; k += 4) {
    v2f a_nxt = a;
    if (k + 4 < K) a_nxt = *(const v2f*)(ap + k + 4);
    v2f b = *(const v2f*)(wp + k);
    acc = __builtin_amdgcn_wmma_f32_16x16x4_f32(false, a, false, b,
                                                (short)0, acc, false, false);
    a = a_nxt;
  }

  const int col  = cb + lrow;
  const float sc = bng[col] * rsqrtf(bnv[col] + LN_EPS);
  const float sh = bnb[col] - bnm[col] * sc;
  const float bz = bias[col];
#pragma unroll
  for (int v = 0; v < 8; ++v) {
    const int row = rb + v + ((lane >> 4) * 8);
    if (row < n) {
      float z = acc[v] + bz;
      out[(long long)row * NOUT + col] = fmaxf(z * sc + sh, 0.0f);
    }
  }
}

// logits[i] = dot(v3[i,:64], w) + b
__global__ void fc4_kernel(const float* __restrict__ v3, const float* __restrict__ w,
                           const float* __restrict__ b, float* __restrict__ out, int n) {
  int i = blockIdx.x * blockDim.x + threadIdx.x;
  if (i >= n) return;
  const float* r = v3 + (long long)i * 64;
  float s = 0.0f;
#pragma unroll
  for (int c = 0; c < 64; c += 4) {
    float4 rv = *(const float4*)(r + c);
    float4 wv = *(const float4*)(w + c);
    s += rv.x * wv.x + rv.y * wv.y + rv.z * wv.z + rv.w * wv.w;
  }
  out[i] = s + b[0];
}

// ---------------------------------------------------------------------------
extern "C" void kernel_launch(void* const* d_in, const int* in_sizes, int n_in,
                              void* d_out, int out_size, void* d_ws, size_t ws_size,
                              hipStream_t stream) {
  (void)in_sizes; (void)n_in; (void)out_size; (void)ws_size;
  // Inputs flattened in setup_inputs() dict insertion order:
  // 0:x  1:edge_index  2:edge_weight
  // per conv l: 3+3l:w_neigh  4+3l:w_root  5+3l:b_root
  // per ln  l: 12+2l:g  13+2l:b
  // clf: 18:fc1.w 19:fc1.b 20..23:bn1(g,b,rm,rv) 24:fc2.w 25:fc2.b 26..29:bn2
  //      30:fc3.w 31:fc3.b 32..35:bn3 36:fc4.w 37:fc4.b
  const float* x   = (const float*)d_in[0];
  const int*   ei  = (const int*)d_in[1];
  const int*   src = ei;
  const int*   dst = ei + N_EDGES;
  const float* ew  = (const float*)d_in[2];
  const float* Wn[3]  = {(const float*)d_in[3],  (const float*)d_in[6],  (const float*)d_in[9]};
  const float* Wr[3]  = {(const float*)d_in[4],  (const float*)d_in[7],  (const float*)d_in[10]};
  const float* Br[3]  = {(const float*)d_in[5],  (const float*)d_in[8],  (const float*)d_in[11]};
  const float* LNg[3] = {(const float*)d_in[12], (const float*)d_in[14], (const float*)d_in[16]};
  const float* LNb[3] = {(const float*)d_in[13], (const float*)d_in[15], (const float*)d_in[17]};
  const float* fcw[3] = {(const float*)d_in[18], (const float*)d_in[24], (const float*)d_in[30]};
  const float* fcb[3] = {(const float*)d_in[19], (const float*)d_in[25], (const float*)d_in[31]};
  const float* bng[3] = {(const float*)d_in[20], (const float*)d_in[26], (const float*)d_in[32]};
  const float* bnb[3] = {(const float*)d_in[21], (const float*)d_in[27], (const float*)d_in[33]};
  const float* bnm[3] = {(const float*)d_in[22], (const float*)d_in[28], (const float*)d_in[34]};
  const float* bnv[3] = {(const float*)d_in[23], (const float*)d_in[29], (const float*)d_in[35]};
  const float* fc4w   = (const float*)d_in[36];
  const float* fc4b   = (const float*)d_in[37];

  // Workspace layout (floats). NH = N*128 = 6.4M.
  //   hA  [0,      NH)      backbone ping (final h lives here)
  //   hB  [NH,    2NH)      backbone pong
  //   agg [2NH,   3NH)      neighbor aggregate; after backbone: v1 spans [NH,3NH)
  //   v2 reuses hA region after fc1; v3 reuses v1 head after fc2
  //   sumw/invw/stats at tail
  const long long NH = (long long)N_NODES * HDIM;
  float* ws    = (float*)d_ws;
  float* hA    = ws;
  float* hB    = ws + NH;
  float* agg   = ws + 2 * NH;
  float* v1    = ws + NH;          // N x 256, overlays hB+agg (free after backbone)
  float* v2    = ws;               // N x 128, overlays hA (free after fc1)
  float* v3    = ws + NH;          // N x 64,  overlays v1 head (free after fc2)
  float* sumw  = ws + 3 * NH;
  float* invw  = sumw + N_NODES;
  float* stats = invw + N_NODES;   // [sum, sumsq, mean, inv_std]

  const int THREADS = 256;
  const int gN    = (N_NODES + THREADS - 1) / THREADS;
  const int gNH   = (int)((NH + THREADS - 1) / THREADS);
  const int gE    = (N_EDGES + THREADS - 1) / THREADS;
  const int gE32  = (int)(((long long)N_EDGES * 32 + THREADS - 1) / THREADS);
  const int gConv = (N_NODES + 127) / 128;
  const int gClf  = (N_NODES + 63) / 64;

  // degree normalization (identical across layers -> once)
  zero_f32<<<gN, THREADS, 0, stream>>>(sumw, N_NODES);
  sumw_kernel<<<gE, THREADS, 0, stream>>>(dst, ew, sumw, N_EDGES);
  invw_kernel<<<gN, THREADS, 0, stream>>>(sumw, invw, N_NODES);

  const float* hin = x;
  float* bufs[2] = {hA, hB};
  for (int l = 0; l < 3; ++l) {
    float* hout = bufs[l & 1];
    zero_f32<<<gNH, THREADS, 0, stream>>>(agg, (int)NH);
    zero_f32<<<1, 32, 0, stream>>>(stats, 4);
    scatter_kernel<<<gE32, THREADS, 0, stream>>>(hin, ew, src, dst, agg, N_EDGES);
    conv_gemm_kernel<<<gConv, 256, 0, stream>>>(agg, invw, hin, Wn[l], Wr[l], Br[l],
                                                hout, stats, N_NODES);
    ln_finalize<<<1, 1, 0, stream>>>(stats, (float)NH);
    ln_apply<<<gNH, THREADS, 0, stream>>>(hout, stats, LNg[l], LNb[l], (int)NH);
    hin = hout;
  }
  // backbone output is in hA (layers: x->hA->hB->hA)

  clf_gemm_kernel<128, 256><<<dim3(gClf, 16), 128, 0, stream>>>(hA, fcw[0], fcb[0],
      bng[0], bnb[0], bnm[0], bnv[0], v1, N_NODES);
  clf_gemm_kernel<256, 128><<<dim3(gClf, 8), 128, 0, stream>>>(v1, fcw[1], fcb[1],
      bng[1], bnb[1], bnm[1], bnv[1], v2, N_NODES);
  clf_gemm_kernel<128, 64><<<dim3(gClf, 4), 128, 0, stream>>>(v2, fcw[2], fcb[2],
      bng[2], bnb[2], bnm[2], bnv[2], v3, N_NODES);
  fc4_kernel<<<gN, THREADS, 0, stream>>>(v3, fc4w, fc4b, (float*)d_out, N_NODES);
}